// TopKRouter_34067680592371
// MI455X (gfx1250) — compile-verified
//
#include <hip/hip_runtime.h>
#include <math.h>

#define T_TOKENS 16384
#define HID      2048
#define NEXP     64
#define TOPK     8
#define NGRP     8
#define EPG      8
#define KGRP     4
#define RSCALE   2.5f
#define REPS     1e-9f

#define BLK_TOK  128   // tokens per block (4 waves x 32 tokens)
#define KC       64    // K-chunk staged in LDS
#define SD       68    // padded LDS row stride (dwords): conflict-free b64 frag loads

typedef float v2f __attribute__((ext_vector_type(2)));
typedef float v4f __attribute__((ext_vector_type(4)));
typedef float v8f __attribute__((ext_vector_type(8)));

__global__ __launch_bounds__(128)
void router_zero_ws(float* ws) {
    ws[threadIdx.x] = 0.0f;   // 128 floats: [0..63]=probSum, [64..127]=fracSum
}

__global__ __launch_bounds__(128)
void router_main(const float* __restrict__ tokens, const float* __restrict__ W,
                 int* __restrict__ out_i, float* __restrict__ out_w,
                 float* __restrict__ gProb, float* __restrict__ gFrac) {
    // LDS: token chunk [128][SD] + weight chunk [64][SD]  (52224 bytes)
    __shared__ __align__(16) float smem[BLK_TOK * SD + NEXP * SD];
    float* sT = smem;                       // [128][SD] staging (GEMM phase)
    float* sW = smem + BLK_TOK * SD;        // [64][SD]  staging (GEMM phase)
    // Epilogue aliases (only touched after the GEMM's final barrier):
    float* sS    = smem;                    // [128][SD] sigmoid scores
    float* sInv  = smem + BLK_TOK * SD;     // [128] 1/rowsum
    float* sFrac = smem + BLK_TOK * SD + 128; // [64] per-block assignment fractions

    const int tid  = threadIdx.x;
    const int lane = tid & 31;
    const int wv   = tid >> 5;        // wave 0..3
    const int m    = lane & 15;       // M/N index within fragment
    const int kh   = lane >> 4;       // K-half select (ISA A/B layout)
    const int tb   = blockIdx.x * BLK_TOK;

    v8f acc[2][4] = {};               // 32 tokens x 64 experts per wave

    for (int kc = 0; kc < HID; kc += KC) {
        // ---- stage token chunk [128][KC] (non-temporal: streamed once) ----
        #pragma unroll
        for (int j = 0; j < 16; ++j) {
            int i = tid + j * 128;
            int r = i >> 4, c = (i & 15) * 4;
            const v4f* src = reinterpret_cast<const v4f*>(
                tokens + (size_t)(tb + r) * HID + kc + c);
            v4f v = __builtin_nontemporal_load(src);
            *reinterpret_cast<v4f*>(&sT[r * SD + c]) = v;
        }
        // ---- stage weight chunk [64][KC] (L2-resident, reused by all blocks) ----
        #pragma unroll
        for (int j = 0; j < 8; ++j) {
            int i = tid + j * 128;
            int r = i >> 4, c = (i & 15) * 4;
            v4f v = *reinterpret_cast<const v4f*>(W + (size_t)r * HID + kc + c);
            *reinterpret_cast<v4f*>(&sW[r * SD + c]) = v;
        }
        __syncthreads();

        // ---- compute: 16 K-steps of V_WMMA_F32_16X16X4_F32 ----
        #pragma unroll 4
        for (int ks = 0; ks < KC; ks += 4) {
            int ko = ks + 2 * kh;
            v2f a0 = *reinterpret_cast<const v2f*>(&sT[(wv * 32 + m)      * SD + ko]);
            v2f a1 = *reinterpret_cast<const v2f*>(&sT[(wv * 32 + 16 + m) * SD + ko]);
            #pragma unroll
            for (int nt = 0; nt < 4; ++nt) {
                v2f b = *reinterpret_cast<const v2f*>(&sW[(nt * 16 + m) * SD + ko]);
                acc[0][nt] = __builtin_amdgcn_wmma_f32_16x16x4_f32(
                    false, a0, false, b, (short)0, acc[0][nt], false, false);
                acc[1][nt] = __builtin_amdgcn_wmma_f32_16x16x4_f32(
                    false, a1, false, b, (short)0, acc[1][nt], false, false);
            }
        }
        __syncthreads();
    }

    // ---- epilogue phase 1: sigmoid, transpose C-layout -> per-token rows in LDS ----
    #pragma unroll
    for (int mt = 0; mt < 2; ++mt)
        #pragma unroll
        for (int nt = 0; nt < 4; ++nt)
            #pragma unroll
            for (int j = 0; j < 8; ++j) {
                float x = acc[mt][nt][j];
                float sg = 1.0f / (1.0f + __expf(-x));
                int tr = wv * 32 + mt * 16 + j + 8 * kh;  // C-layout: M = j + 8*(lane>=16)
                int e  = nt * 16 + m;                     // N = lane%16
                sS[tr * SD + e] = sg;
            }
    if (tid < NEXP) sFrac[tid] = 0.0f;
    __syncthreads();

    // ---- epilogue phase 2: per-token routing (thread t owns token tb+t) ----
    {
        float tot = 0.0f;
        unsigned key[NEXP];
        #pragma unroll
        for (int e = 0; e < NEXP; ++e) {
            float v = sS[tid * SD + e];
            tot += v;
            key[e] = (__float_as_uint(v) & 0xFFFFFFC0u) | (unsigned)e; // positive floats: bit order == value order
        }
        // group maxima + top-4 groups
        unsigned gk[NGRP];
        #pragma unroll
        for (int g = 0; g < NGRP; ++g) {
            unsigned b = 0u;
            #pragma unroll
            for (int q = 0; q < EPG; ++q) b = max(b, key[g * EPG + q]);
            gk[g] = b;
        }
        unsigned gsel = 0u;
        #pragma unroll
        for (int p = 0; p < KGRP; ++p) {
            unsigned best = 0u;
            #pragma unroll
            for (int g = 0; g < NGRP; ++g) best = max(best, gk[g]);
            gsel |= 1u << ((best & 63u) >> 3);
            #pragma unroll
            for (int g = 0; g < NGRP; ++g) gk[g] = (gk[g] == best) ? 0u : gk[g];
        }
        // mask experts outside selected groups
        #pragma unroll
        for (int e = 0; e < NEXP; ++e)
            key[e] = ((gsel >> (e >> 3)) & 1u) ? key[e] : 0u;
        // top-8 selection
        int   idx[TOPK];
        float wvv[TOPK];
        float wsum = 0.0f;
        #pragma unroll
        for (int p = 0; p < TOPK; ++p) {
            unsigned best = 0u;
            #pragma unroll
            for (int e = 0; e < NEXP; ++e) best = max(best, key[e]);
            int e = (int)(best & 63u);
            float v = sS[tid * SD + e];        // exact score
            idx[p] = e; wvv[p] = v; wsum += v;
            #pragma unroll
            for (int e2 = 0; e2 < NEXP; ++e2) key[e2] = (key[e2] == best) ? 0u : key[e2];
        }
        float scl = RSCALE / fmaxf(wsum, REPS);
        int tok = tb + tid;
        #pragma unroll
        for (int p = 0; p < TOPK; ++p) {
            out_i[tok * TOPK + p] = idx[p];
            out_w[tok * TOPK + p] = wvv[p] * scl;
            atomicAdd(&sFrac[idx[p]], 0.125f);  // assignments.mean(axis=1): 1/K each
        }
        sInv[tid] = 1.0f / fmaxf(tot, REPS);
    }
    __syncthreads();

    // ---- epilogue phase 3: per-expert column sums of normalized scores ----
    if (tid < NEXP) {
        float cs = 0.0f;
        for (int r = 0; r < BLK_TOK; ++r)
            cs += sS[r * SD + tid] * sInv[r];
        atomicAdd(&gProb[tid], cs);
        atomicAdd(&gFrac[tid], sFrac[tid]);
    }
}

__global__ void router_finalize(const float* __restrict__ ws, float* __restrict__ aux) {
    if (threadIdx.x == 0) {
        float s = 0.0f;
        for (int e = 0; e < NEXP; ++e) s += ws[e] * ws[NEXP + e];
        *aux = s * (float)NEXP / ((float)T_TOKENS * (float)T_TOKENS);
    }
}

extern "C" void kernel_launch(void* const* d_in, const int* in_sizes, int n_in,
                              void* d_out, int out_size, void* d_ws, size_t ws_size,
                              hipStream_t stream) {
    const float* tokens = (const float*)d_in[0];
    const float* W      = (const float*)d_in[1];
    int*   out_i = (int*)d_out;                              // [16384, 8] int32 bits
    float* out_w = (float*)d_out + T_TOKENS * TOPK;          // [16384, 8] f32
    float* aux   = (float*)d_out + 2 * T_TOKENS * TOPK;      // scalar
    float* ws    = (float*)d_ws;                             // [64] prob + [64] frac

    router_zero_ws<<<1, 128, 0, stream>>>(ws);
    router_main<<<T_TOKENS / BLK_TOK, 128, 0, stream>>>(tokens, W, out_i, out_w,
                                                        ws, ws + NEXP);
    router_finalize<<<1, 32, 0, stream>>>(ws, aux);
}